// SSIMLoss_15848429322438
// MI455X (gfx1250) — compile-verified
//
#include <hip/hip_runtime.h>
#include <hip/hip_bf16.h>

// SSIM loss, 32x1x512x512 fp32, 11x11 Gaussian (separable) window.
// Row+col convolution passes mapped onto V_WMMA_F32_16X16X4_F32 (fp32 WMMA),
// SSIM map + mean fused, deterministic two-stage reduction.
//
// Round-1 fixes: branch-free 5-signal horizontal pass (switch removed),
// job-invariant banded weights hoisted to registers, wave id made
// SGPR-uniform via readfirstlane so WMMA regions run under full EXEC.

typedef __attribute__((ext_vector_type(2))) float v2f;
typedef __attribute__((ext_vector_type(8))) float v8f;

#define IMG_H 512
#define IMG_W 512
#define IMG_B 32
#define TILE 32            // output tile edge per workgroup
#define HALO 5             // (11-1)/2
#define IN_EDGE (TILE + 2*HALO)   // 42 valid input rows/cols
#define IROWS 48           // padded to 3x16 row-tiles for WMMA
#define ISTR  45           // input LDS row stride (>= 44 needed by K-padding reads)
#define HROWS 48
#define HSTR  33           // hconv LDS row stride (32 + 1 pad)
#define NTX (IMG_W / TILE) // 16
#define NTY (IMG_H / TILE) // 16
#define NWG (NTX * NTY * IMG_B)   // 8192 workgroups
#define NPIX ((float)(IMG_B * IMG_H * IMG_W)) // 8388608

static __device__ inline v8f wmma4(v2f a, v2f b, v8f c) {
  // V_WMMA_F32_16X16X4_F32 : D[16x16] += A[16x4] * B[4x16], all fp32
  return __builtin_amdgcn_wmma_f32_16x16x4_f32(
      /*neg_a=*/false, a, /*neg_b=*/false, b,
      /*c_mod=*/(short)0, c, /*reuse_a=*/false, /*reuse_b=*/false);
}

// Banded Gaussian weight: w[k-n] if 0 <= k-n <= 10 else 0  (LDS-resident w)
static __device__ inline float bandw(const float* __restrict__ sw, int k, int n) {
  int d = k - n;
  bool ok = (d >= 0) && (d <= 10);
  float v = sw[ok ? d : 0];
  return ok ? v : 0.0f;
}

__global__ __launch_bounds__(128)
void ssim_tile_kernel(const float* __restrict__ pred,
                      const float* __restrict__ targ,
                      const float* __restrict__ window, // [1,1,11,11]
                      float* __restrict__ partials) {
  __shared__ float sP[IROWS * ISTR];
  __shared__ float sT[IROWS * ISTR];
  __shared__ float sH[5][HROWS * HSTR];
  __shared__ float sW[11];
  __shared__ float sRed[128];

  const int tid  = threadIdx.x;
  const int lane = tid & 31;
  const int half = lane >> 4;   // 0: lanes 0-15 (K 0,1) ; 1: lanes 16-31 (K 2,3)
  const int mrow = lane & 15;   // M (A rows) or N (B cols / D cols)
  // wave index is uniform across the wave: pin it to an SGPR so job loops are
  // scalar-controlled and WMMA executes under full EXEC.
  const int wave = __builtin_amdgcn_readfirstlane(tid >> 5);

  const int bx = blockIdx.x, by = blockIdx.y, bz = blockIdx.z;

  // ---- 1-D Gaussian from the 2D window: win[5][j] = w5*w[j], w5 = sqrt(win[5][5])
  if (tid < 11) sW[tid] = window[55 + tid] / sqrtf(window[60]);

  // ---- zero-fill halo padding region, then load valid 42x42 tile (SAME padding)
  for (int i = tid; i < IROWS * ISTR; i += 128) { sP[i] = 0.0f; sT[i] = 0.0f; }
  __syncthreads();
  for (int i = tid; i < IN_EDGE * IN_EDGE; i += 128) {
    int r = i / IN_EDGE, c = i % IN_EDGE;
    int gy = by * TILE + r - HALO;
    int gx = bx * TILE + c - HALO;
    float pv = 0.0f, tv = 0.0f;
    if (gy >= 0 && gy < IMG_H && gx >= 0 && gx < IMG_W) {
      size_t g = ((size_t)bz * IMG_H + gy) * IMG_W + gx;
      pv = pred[g];
      tv = targ[g];
    }
    sP[r * ISTR + c] = pv;
    sT[r * ISTR + c] = tv;
  }

  // ---- Job-invariant banded weights, one v2f per K-chunk per lane.
  // Serves as B for the horizontal pass and A for the vertical pass:
  // wb[kk] = { w[k0 - mrow], w[k0+1 - mrow] }, k0 = kk*4 + half*2.
  v2f wb[7];
#pragma unroll
  for (int kk = 0; kk < 7; ++kk) {
    int k0 = kk * 4 + half * 2;
    wb[kk].x = bandw(sW, k0,     mrow);
    wb[kk].y = bandw(sW, k0 + 1, mrow);
  }
  __syncthreads();

  // ---- Stage 2: horizontal conv via WMMA, all 5 signals per position.
  // out_h[y][x0+n] = sum_k sig[y][x0+k] * w[k-n], k = 0..25 (7 chunks of K=4).
  // 6 positions = 3 row-tiles x 2 col-tiles, round-robin over 4 waves.
  for (int pos = wave; pos < 6; pos += 4) {
    const int ybase = (pos >> 1) * 16;
    const int x0    = (pos & 1) * 16;
    const int arow  = ybase + mrow;
    v8f acc0 = {0.f,0.f,0.f,0.f,0.f,0.f,0.f,0.f};
    v8f acc1 = acc0, acc2 = acc0, acc3 = acc0, acc4 = acc0;
#pragma unroll
    for (int kk = 0; kk < 7; ++kk) {
      const int c0 = x0 + kk * 4 + half * 2;
      const float p0 = sP[arow * ISTR + c0], p1 = sP[arow * ISTR + c0 + 1];
      const float t0 = sT[arow * ISTR + c0], t1 = sT[arow * ISTR + c0 + 1];
      v2f ap  = {p0,      p1};
      v2f at  = {t0,      t1};
      v2f app = {p0 * p0, p1 * p1};
      v2f att = {t0 * t0, t1 * t1};
      v2f apt = {p0 * t0, p1 * t1};
      acc0 = wmma4(ap,  wb[kk], acc0);
      acc1 = wmma4(at,  wb[kk], acc1);
      acc2 = wmma4(app, wb[kk], acc2);
      acc3 = wmma4(att, wb[kk], acc3);
      acc4 = wmma4(apt, wb[kk], acc4);
    }
    // D layout: reg r holds M = r + half*8, N = mrow
#pragma unroll
    for (int r = 0; r < 8; ++r) {
      const int o = (ybase + half * 8 + r) * HSTR + (x0 + mrow);
      sH[0][o] = acc0[r];
      sH[1][o] = acc1[r];
      sH[2][o] = acc2[r];
      sH[3][o] = acc3[r];
      sH[4][o] = acc4[r];
    }
  }
  __syncthreads();

  // ---- Stage 3: vertical conv via WMMA + fused SSIM map.
  // out[y0+m][x0+n] = sum_k w[k-m] * hconv[y0+k][x0+n]; one 16x16 position per wave.
  const int y0 = (wave >> 1) * 16;
  const int x0 = (wave & 1) * 16;
  v8f vac0 = {0.f,0.f,0.f,0.f,0.f,0.f,0.f,0.f};
  v8f vac1 = vac0, vac2 = vac0, vac3 = vac0, vac4 = vac0;

#pragma unroll
  for (int kk = 0; kk < 7; ++kk) {
    const int k0 = kk * 4 + half * 2;
    const int o0 = (y0 + k0)     * HSTR + x0 + mrow;
    const int o1 = (y0 + k0 + 1) * HSTR + x0 + mrow;
    v2f b0 = {sH[0][o0], sH[0][o1]};
    v2f b1 = {sH[1][o0], sH[1][o1]};
    v2f b2 = {sH[2][o0], sH[2][o1]};
    v2f b3 = {sH[3][o0], sH[3][o1]};
    v2f b4 = {sH[4][o0], sH[4][o1]};
    vac0 = wmma4(wb[kk], b0, vac0);
    vac1 = wmma4(wb[kk], b1, vac1);
    vac2 = wmma4(wb[kk], b2, vac2);
    vac3 = wmma4(wb[kk], b3, vac3);
    vac4 = wmma4(wb[kk], b4, vac4);
  }

  // SSIM map per element (8 output pixels per lane) + local sum
  const float C1 = 0.0001f, C2 = 0.0009f;
  float lsum = 0.0f;
#pragma unroll
  for (int r = 0; r < 8; ++r) {
    float mu1 = vac0[r], mu2 = vac1[r];
    float ep2 = vac2[r], et2 = vac3[r], ept = vac4[r];
    float mu1s = mu1 * mu1, mu2s = mu2 * mu2, m12 = mu1 * mu2;
    float s1 = ep2 - mu1s, s2 = et2 - mu2s, s12 = ept - m12;
    float num = (2.0f * m12 + C1) * (2.0f * s12 + C2);
    float den = (mu1s + mu2s + C1) * (s1 + s2 + C2);
    lsum += num / den;
  }

  // ---- deterministic in-block tree reduction -> one partial per workgroup
  sRed[tid] = lsum;
  __syncthreads();
#pragma unroll
  for (int st = 64; st > 0; st >>= 1) {
    if (tid < st) sRed[tid] += sRed[tid + st];
    __syncthreads();
  }
  if (tid == 0) {
    int wg = (bz * NTY + by) * NTX + bx;
    partials[wg] = sRed[0];
  }
}

__global__ __launch_bounds__(256)
void ssim_reduce_kernel(const float* __restrict__ partials,
                        float* __restrict__ out) {
  __shared__ float sr[256];
  float s = 0.0f;
  for (int i = threadIdx.x; i < NWG; i += 256) s += partials[i];
  sr[threadIdx.x] = s;
  __syncthreads();
#pragma unroll
  for (int st = 128; st > 0; st >>= 1) {
    if (threadIdx.x < st) sr[threadIdx.x] += sr[threadIdx.x + st];
    __syncthreads();
  }
  if (threadIdx.x == 0) out[0] = 1.0f - sr[0] / NPIX;
}

extern "C" void kernel_launch(void* const* d_in, const int* in_sizes, int n_in,
                              void* d_out, int out_size, void* d_ws, size_t ws_size,
                              hipStream_t stream) {
  const float* pred   = (const float*)d_in[0];
  const float* target = (const float*)d_in[1];
  const float* window = (const float*)d_in[2];
  float* partials = (float*)d_ws;   // NWG floats = 32 KB of scratch
  float* out = (float*)d_out;

  dim3 grid(NTX, NTY, IMG_B);
  ssim_tile_kernel<<<grid, 128, 0, stream>>>(pred, target, window, partials);
  ssim_reduce_kernel<<<1, 256, 0, stream>>>(partials, out);
}